// SparseGCNConv_58411555225975
// MI455X (gfx1250) — compile-verified
//
#include <hip/hip_runtime.h>
#include <hip/hip_bf16.h>

#define N_NODES_C     100000
#define IN_FEATS      256
#define OUT_FEATS     256
#define NODES_PER_BLK 16
#define A_STRIDE      260   // padded LDS row stride in floats (bank spread, 8B aligned)

typedef float v2f __attribute__((ext_vector_type(2)));
typedef float v4f __attribute__((ext_vector_type(4)));
typedef float v8f __attribute__((ext_vector_type(8)));

// Pre-pass: Wt[k][o] = W[o][k]  (256x256 fp32, 256 KB, lives in d_ws; hot in L2)
__global__ void SparseGCNConv_transposeW(const float* __restrict__ W,
                                         float* __restrict__ Wt) {
    const int o = blockIdx.x;   // 0..255 output feature
    const int k = threadIdx.x;  // 0..255 input feature
    Wt[k * OUT_FEATS + o] = __builtin_nontemporal_load(&W[o * IN_FEATS + k]);
}

// Fused kernel: per block, aggregate 16 destination nodes into LDS (SpMM via
// sorted-dst binary search, no global atomics, no HBM round-trip for agg),
// then 16x256 GEMM tile with V_WMMA_F32_16X16X4_F32 and bias add.
__global__ __launch_bounds__(256)
void SparseGCNConv_fused(const float* __restrict__ feat,   // [N, 256] L2-resident
                         const int*   __restrict__ esrc,   // [E] read-once -> NT
                         const int*   __restrict__ edst,   // [E] sorted
                         const float* __restrict__ ew,     // [E] read-once -> NT
                         const float* __restrict__ Wt,     // [256, 256] = W^T
                         const float* __restrict__ bias,   // [256]
                         float*       __restrict__ out,    // [N, 256] streamed -> NT
                         int n_edges) {
    __shared__ float Alds[NODES_PER_BLK * A_STRIDE];
    __shared__ int   bounds[NODES_PER_BLK + 1];

    const int tid  = threadIdx.x;
    const int base = blockIdx.x * NODES_PER_BLK;

    // --- zero the A tile (ds_add accumulation target) while threads 0..16
    //     run the segment-boundary binary searches (disjoint LDS arrays) -----
    for (int i = tid; i < NODES_PER_BLK * A_STRIDE; i += 256) Alds[i] = 0.0f;
    if (tid <= NODES_PER_BLK) {
        const int target = base + tid;
        int lo = 0, hi = n_edges;
        while (lo < hi) {
            const int mid = (lo + hi) >> 1;
            if (edst[mid] < target) lo = mid + 1; else hi = mid;
        }
        bounds[tid] = lo;
    }
    __syncthreads();

    // --- phase 1: aggregate 16 node rows into LDS ---------------------------
    // 64 column-threads x float4 => global_load_b128, full-cacheline requests;
    // 4 edge-groups run 4 edges concurrently to hide L2 latency. Partials are
    // combined with LDS float atomics (ds_add_f32). Edge streams are marked
    // non-temporal so the 102 MB feature table stays resident in the 192 MB L2.
    {
        const int grp = tid >> 6;          // 0..3 : which edge of every 4
        const int ci  = (tid & 63) * 4;    // float4 column base, 0..252
        for (int n = 0; n < NODES_PER_BLK; ++n) {
            const int lo = bounds[n];
            const int hi = bounds[n + 1];
            float ax = 0.0f, ay = 0.0f, az = 0.0f, aw = 0.0f;
            for (int e = lo + grp; e < hi; e += 4) {
                const int   s = __builtin_nontemporal_load(&esrc[e]);
                const float w = __builtin_nontemporal_load(&ew[e]);
                const v4f f = *(const v4f*)(feat + (size_t)s * IN_FEATS + ci);
                ax = fmaf(f.x, w, ax);
                ay = fmaf(f.y, w, ay);
                az = fmaf(f.z, w, az);
                aw = fmaf(f.w, w, aw);
            }
            float* dst = &Alds[n * A_STRIDE + ci];
            atomicAdd(dst + 0, ax);
            atomicAdd(dst + 1, ay);
            atomicAdd(dst + 2, az);
            atomicAdd(dst + 3, aw);
        }
    }
    __syncthreads();   // orders all ds_add_f32 before the WMMA reads

    // --- phase 2: out[base:base+16, :] = Alds @ W^T + bias via WMMA ---------
    const int wave = tid >> 5;         // 8 waves
    const int lane = tid & 31;
    const int half = lane >> 4;        // 0: lanes 0-15, 1: lanes 16-31
    const int l16  = lane & 15;
    const int col0 = wave * 32;        // each wave owns 2 adjacent 16-col tiles

    v8f acc0 = {};
    v8f acc1 = {};

    for (int kk = 0; kk < IN_FEATS; kk += 4) {
        const int ka = kk + half * 2;  // K pair handled by this half-wave

        // A fragment (16x4): half 0 holds K={kk,kk+1}, half 1 K={kk+2,kk+3}
        v2f a;
        a.x = Alds[l16 * A_STRIDE + ka];
        a.y = Alds[l16 * A_STRIDE + ka + 1];

        // B fragments (4x16), B[k][n] = Wt[k*256 + n] -> lane-contiguous loads
        v2f b0, b1;
        b0.x = Wt[(ka + 0) * OUT_FEATS + col0 + l16];
        b0.y = Wt[(ka + 1) * OUT_FEATS + col0 + l16];
        b1.x = Wt[(ka + 0) * OUT_FEATS + col0 + 16 + l16];
        b1.y = Wt[(ka + 1) * OUT_FEATS + col0 + 16 + l16];

        acc0 = __builtin_amdgcn_wmma_f32_16x16x4_f32(
            false, a, false, b0, (short)0, acc0, false, false);
        acc1 = __builtin_amdgcn_wmma_f32_16x16x4_f32(
            false, a, false, b1, (short)0, acc1, false, false);
    }

    // --- store: C/D layout -> VGPR j is row M=j (lanes 0-15) / M=j+8 (16-31)
    // Non-temporal: output is write-once, keep it out of L2.
    const float bv0 = bias[col0 + l16];
    const float bv1 = bias[col0 + 16 + l16];
    const int orow0 = base + half * 8;
    for (int j = 0; j < 8; ++j) {
        const size_t r = (size_t)(orow0 + j) * OUT_FEATS;
        __builtin_nontemporal_store(acc0[j] + bv0, &out[r + col0 + l16]);
        __builtin_nontemporal_store(acc1[j] + bv1, &out[r + col0 + 16 + l16]);
    }
}

extern "C" void kernel_launch(void* const* d_in, const int* in_sizes, int n_in,
                              void* d_out, int out_size, void* d_ws, size_t ws_size,
                              hipStream_t stream) {
    const float* feat = (const float*)d_in[0];   // [100000, 256]
    const int*   esrc = (const int*)d_in[1];     // [E]
    const int*   edst = (const int*)d_in[2];     // [E] sorted
    const float* ew   = (const float*)d_in[3];   // [E]
    const float* W    = (const float*)d_in[4];   // [256, 256]
    const float* bias = (const float*)d_in[5];   // [256]
    float*       out  = (float*)d_out;           // [100000, 256]
    float*       Wt   = (float*)d_ws;            // 256 KB scratch

    const int n_edges = in_sizes[1];

    SparseGCNConv_transposeW<<<OUT_FEATS, IN_FEATS, 0, stream>>>(W, Wt);

    const int n_blocks = N_NODES_C / NODES_PER_BLK;  // 6250
    SparseGCNConv_fused<<<n_blocks, 256, 0, stream>>>(
        feat, esrc, edst, ew, Wt, bias, out, n_edges);
}